// GATLayer_46497315946703
// MI455X (gfx1250) — compile-verified
//
#include <hip/hip_runtime.h>
#include <math.h>

#define N_NODES   100000
#define N_EDGES   3200000
#define F         256
#define NEG_SLOPE 0.2f

typedef __attribute__((ext_vector_type(2))) float v2f;
typedef __attribute__((ext_vector_type(8))) float v8f;

// ---------------------------------------------------------------------------
// init: zero output accumulator, set per-node max = -inf, sum = 0
// ---------------------------------------------------------------------------
__global__ void gat_init_kernel(float* __restrict__ out,
                                float* __restrict__ mmax,
                                float* __restrict__ ssum) {
    int gid = blockIdx.x * blockDim.x + threadIdx.x;   // 25.6M threads exactly
    out[gid] = 0.0f;
    if (gid < N_NODES) {
        mmax[gid] = -INFINITY;
        ssum[gid] = 0.0f;
    }
}

// ---------------------------------------------------------------------------
// h = x @ Wv + bv  via V_WMMA_F32_16X16X4_F32 (fp32 A/B, fp32 accum)
// One wave computes a 16(M) x 64(N) strip: 4 accumulators, A reused 4x.
// Layouts (ISA 7.12.2, wave32):
//   A 16x4 : lane(0..15)  v2f = {A[m=lane][k0+0], A[m=lane][k0+1]}
//            lane(16..31) v2f = {A[m=lane-16][k0+2], A[m=lane-16][k0+3]}
//   B 4x16 : mirrored over N
//   C/D    : VGPR r, lanes 0-15 -> M=r,   N=lane
//                    lanes 16-31 -> M=r+8, N=lane-16
// ---------------------------------------------------------------------------
__global__ void gat_gemm_wmma_kernel(const float* __restrict__ x,
                                     const float* __restrict__ Wv,
                                     const float* __restrict__ bv,
                                     float* __restrict__ h) {
    const int wave = blockIdx.x * (blockDim.x >> 5) + (threadIdx.x >> 5);
    const int lane = threadIdx.x & 31;
    const int half = lane >> 4;     // 0 or 1
    const int lm   = lane & 15;     // row (A) / col (B,C,D) within 16-tile

    const int mtile = wave >> 2;    // 0..6249
    const int ngrp  = wave & 3;     // 0..3
    const int m0 = mtile * 16;
    const int n0 = ngrp * 64;

    v8f acc0 = {}, acc1 = {}, acc2 = {}, acc3 = {};

    const float* arow = x + (size_t)(m0 + lm) * F + 2 * half;

    for (int k0 = 0; k0 < F; k0 += 4) {
        v2f a;
        a.x = arow[k0];
        a.y = arow[k0 + 1];

        const float* b0 = Wv + (size_t)(k0 + 2 * half) * F + n0 + lm;
        const float* b1 = b0 + F;
        v2f b;

        b.x = b0[0];  b.y = b1[0];
        acc0 = __builtin_amdgcn_wmma_f32_16x16x4_f32(false, a, false, b,
                                                     (short)0, acc0, false, false);
        b.x = b0[16]; b.y = b1[16];
        acc1 = __builtin_amdgcn_wmma_f32_16x16x4_f32(false, a, false, b,
                                                     (short)0, acc1, false, false);
        b.x = b0[32]; b.y = b1[32];
        acc2 = __builtin_amdgcn_wmma_f32_16x16x4_f32(false, a, false, b,
                                                     (short)0, acc2, false, false);
        b.x = b0[48]; b.y = b1[48];
        acc3 = __builtin_amdgcn_wmma_f32_16x16x4_f32(false, a, false, b,
                                                     (short)0, acc3, false, false);
    }

    const int mbase = m0 + 8 * half;
    const int n = n0 + lm;
#pragma unroll
    for (int r = 0; r < 8; ++r) {
        float* hrow = h + (size_t)(mbase + r) * F;
        hrow[n]      = acc0[r] + bv[n];
        hrow[n + 16] = acc1[r] + bv[n + 16];
        hrow[n + 32] = acc2[r] + bv[n + 32];
        hrow[n + 48] = acc3[r] + bv[n + 48];
    }
}

// ---------------------------------------------------------------------------
// q = h @ wq + bq ; k = h @ wk + bk   (one wave per node, dual dot product)
// ---------------------------------------------------------------------------
__global__ void gat_qk_kernel(const float* __restrict__ h,
                              const float* __restrict__ wq,
                              const float* __restrict__ bq,
                              const float* __restrict__ wk,
                              const float* __restrict__ bk,
                              float* __restrict__ q,
                              float* __restrict__ k) {
    const int node = blockIdx.x * (blockDim.x >> 5) + (threadIdx.x >> 5);
    const int lane = threadIdx.x & 31;
    if (node >= N_NODES) return;

    const float* hrow = h + (size_t)node * F;
    float qs = 0.0f, ks = 0.0f;
#pragma unroll
    for (int j = 0; j < F / 32; ++j) {
        float hv = hrow[lane + j * 32];
        qs += hv * wq[lane + j * 32];
        ks += hv * wk[lane + j * 32];
    }
#pragma unroll
    for (int off = 16; off > 0; off >>= 1) {
        qs += __shfl_xor(qs, off, 32);
        ks += __shfl_xor(ks, off, 32);
    }
    if (lane == 0) {
        q[node] = qs + bq[0];
        k[node] = ks + bk[0];
    }
}

// ---------------------------------------------------------------------------
// float atomic max via signed/unsigned ordering trick (works for all floats
// given the slot is initialized to -inf)
// ---------------------------------------------------------------------------
__device__ inline void atomicMaxFloat(float* addr, float value) {
    if (value >= 0.0f) {
        atomicMax((int*)addr, __float_as_int(value));
    } else {
        atomicMin((unsigned int*)addr, __float_as_uint(value));
    }
}

__device__ inline float leaky(float e) {
    return e > 0.0f ? e : NEG_SLOPE * e;
}

// per-edge: segment max of leaky-relu'd logits over dst
__global__ void gat_edge_max_kernel(const int* __restrict__ src,
                                    const int* __restrict__ dst,
                                    const float* __restrict__ q,
                                    const float* __restrict__ k,
                                    float* __restrict__ mmax) {
    int e = blockIdx.x * blockDim.x + threadIdx.x;   // exact: 3.2M
    int d = dst[e];
    float c = leaky(q[src[e]] + k[d]);
    atomicMaxFloat(&mmax[d], c);
}

// per-edge: segment sum of exp(c - max)
__global__ void gat_edge_sum_kernel(const int* __restrict__ src,
                                    const int* __restrict__ dst,
                                    const float* __restrict__ q,
                                    const float* __restrict__ k,
                                    const float* __restrict__ mmax,
                                    float* __restrict__ ssum) {
    int e = blockIdx.x * blockDim.x + threadIdx.x;
    int d = dst[e];
    float c = leaky(q[src[e]] + k[d]);
    atomicAdd(&ssum[d], __expf(c - mmax[d]));
}

// ---------------------------------------------------------------------------
// scatter: out[dst] += h[src] * attn      (one wave per edge, 8 feats/lane)
// h and out both fit in the 192MB L2, so gathers + atomics stay on-chip.
// ---------------------------------------------------------------------------
__global__ void gat_scatter_kernel(const int* __restrict__ src,
                                   const int* __restrict__ dst,
                                   const float* __restrict__ q,
                                   const float* __restrict__ k,
                                   const float* __restrict__ mmax,
                                   const float* __restrict__ ssum,
                                   const float* __restrict__ h,
                                   float* __restrict__ out) {
    const int e    = blockIdx.x * (blockDim.x >> 5) + (threadIdx.x >> 5); // exact
    const int lane = threadIdx.x & 31;

    const int s = src[e];
    const int d = dst[e];
    float c = leaky(q[s] + k[d]);
    float attn = __expf(c - mmax[d]) / ssum[d];

    const float* hrow = h + (size_t)s * F;
    float* orow = out + (size_t)d * F;
#pragma unroll
    for (int j = 0; j < F / 32; ++j) {
        int f = lane + j * 32;
        atomicAdd(&orow[f], hrow[f] * attn);
    }
}

// ---------------------------------------------------------------------------
// launch
// ---------------------------------------------------------------------------
extern "C" void kernel_launch(void* const* d_in, const int* in_sizes, int n_in,
                              void* d_out, int out_size, void* d_ws, size_t ws_size,
                              hipStream_t stream) {
    const float* x  = (const float*)d_in[0];   // [N, 256]
    const float* Wv = (const float*)d_in[1];   // [256, 256]
    const float* bv = (const float*)d_in[2];   // [256]
    const float* wq = (const float*)d_in[3];   // [256]
    const float* bq = (const float*)d_in[4];   // scalar
    const float* wk = (const float*)d_in[5];   // [256]
    const float* bk = (const float*)d_in[6];   // scalar
    const int*   src = (const int*)d_in[7];    // [E]
    const int*   dst = (const int*)d_in[8];    // [E]
    float* out = (float*)d_out;                // [N, 256]

    // workspace layout
    char* ws = (char*)d_ws;
    float* h    = (float*)ws;                         ws += (size_t)N_NODES * F * sizeof(float);
    float* q    = (float*)ws;                         ws += (size_t)N_NODES * sizeof(float);
    float* k    = (float*)ws;                         ws += (size_t)N_NODES * sizeof(float);
    float* mmax = (float*)ws;                         ws += (size_t)N_NODES * sizeof(float);
    float* ssum = (float*)ws;

    // 1) init out / max / sum          (N*F = 25.6M threads exactly)
    gat_init_kernel<<<(N_NODES * F) / 256, 256, 0, stream>>>(out, mmax, ssum);

    // 2) h = x@Wv + bv  via WMMA fp32  (25000 waves = 3125 blocks of 8 waves)
    gat_gemm_wmma_kernel<<<3125, 256, 0, stream>>>(x, Wv, bv, h);

    // 3) q,k per node                  (8 waves/block -> 12500 blocks)
    gat_qk_kernel<<<(N_NODES + 7) / 8, 256, 0, stream>>>(h, wq, bq, wk, bk, q, k);

    // 4) segment max over dst
    gat_edge_max_kernel<<<N_EDGES / 256, 256, 0, stream>>>(src, dst, q, k, mmax);

    // 5) segment sum of exp
    gat_edge_sum_kernel<<<N_EDGES / 256, 256, 0, stream>>>(src, dst, q, k, mmax, ssum);

    // 6) weighted scatter-add          (one wave per edge -> 400000 blocks)
    gat_scatter_kernel<<<N_EDGES / 8, 256, 0, stream>>>(src, dst, q, k, mmax, ssum, h, out);
}